// ROIAveragePool_25752623907188
// MI455X (gfx1250) — compile-verified
//
#include <hip/hip_runtime.h>
#include <hip/hip_bf16.h>
#include <stdint.h>

// ---------------------------------------------------------------------------
// ROI average pool via 2D integral image (summed-area table).
//
// Roofline: after the SAT transform the problem is pure data movement:
//   pass1: 64MB R + 64MB W (HBM streaming)
//   pass2: 64MB R + 64MB W (L2-resident: 128MB < 192MB L2)
//   lookup: ~8MB gathered R + 2MB W
// ~266MB total -> ~11us at 23.3 TB/s. FLOPs are negligible (1 add/elem/pass),
// so WMMA does not apply; the CDNA5-specific path used instead is the Tensor
// Data Mover (tensor_load_to_lds + s_wait_tensorcnt) for pass 2's strided 2D
// tiles, plus global_prefetch in the streaming pass-1 scan.
// ---------------------------------------------------------------------------

#define HH 256
#define WW 256
#define DD 256
#define WD (WW * DD)          // elements per h-row = 65536
#define N_BOXES 2048
#define HC 32                 // h-chunk rows per TDM tile in pass 2

typedef unsigned int u32x4 __attribute__((ext_vector_type(4)));
typedef int          i32x8 __attribute__((ext_vector_type(8)));
typedef int          i32x4 __attribute__((ext_vector_type(4)));

// Build a Tensor DMA Descriptor (D#) for a 2D global->LDS tile load and issue
// TENSOR_LOAD_TO_LDS. Layout per CDNA5 ISA §8.3 (group0) / §8.4 (group1).
//   tile: tile_d1 rows of tile_d0 elements (4B each), rows separated by
//   stride0 elements in global memory; stored densely in LDS at lds_off.
// This toolchain (clang-23 / therock-10.0) exposes the 6-arg builtin:
//   (v4u32 g0, v8i32 g1, v4i32 g2, v4i32 g3, v8i32 g4, i32 cpol)
// g2/g3/g4 carry dims 2-4 state, unused for a 2D tile -> zero-filled.
__device__ __forceinline__ void tdm_load_tile_2d(uint32_t lds_off,
                                                 const void* gptr,
                                                 uint32_t tensor_d0,
                                                 uint32_t tensor_d1,
                                                 uint32_t tile_d0,
                                                 uint32_t tile_d1,
                                                 uint64_t stride0_elems) {
  const uint64_t ga = (uint64_t)(uintptr_t)gptr;
  u32x4 g0;
  // group0: count=1 (valid user descriptor), lds_addr, global_addr[56:0], type=2
  g0[0] = 1u;
  g0[1] = lds_off;
  g0[2] = (uint32_t)ga;
  g0[3] = ((uint32_t)(ga >> 32) & 0x01FFFFFFu) | (2u << 30);

  i32x8 g1;
  // group1: wg_mask=0 (not in cluster), data_size=2 (4 bytes), no barrier/
  // iterate/pad/timeout; tensor dims / tile dims / dim0 stride.
  g1[0] = (int)(2u << 16);
  g1[1] = (int)((tensor_d0 & 0xFFFFu) << 16);                       // td0[15:0] @ [63:48]
  g1[2] = (int)(((tensor_d0 >> 16) & 0xFFFFu) |
                ((tensor_d1 & 0xFFFFu) << 16));                     // td0[31:16] | td1[15:0]
  g1[3] = (int)(((tensor_d1 >> 16) & 0xFFFFu) |
                ((tile_d0 & 0xFFFFu) << 16));                       // td1[31:16] | tile0
  g1[4] = (int)(tile_d1 & 0xFFFFu);                                 // tile1 | tile2=0
  g1[5] = (int)(uint32_t)(stride0_elems & 0xFFFFFFFFu);             // stride0[31:0]
  g1[6] = (int)(uint32_t)((stride0_elems >> 32) & 0xFFFFu);         // stride0[47:32] | t1s lo=0
  g1[7] = 0;

  i32x4 g2 = {0, 0, 0, 0};               // dims 2/3 unused (tile dims zero)
  i32x4 g3 = {0, 0, 0, 0};
  i32x8 g4 = {0, 0, 0, 0, 0, 0, 0, 0};   // extra group (unused for 2D tile)

  __builtin_amdgcn_tensor_load_to_lds(g0, g1, g2, g3, g4, /*cpol=*/0);
}

// ---------------------------------------------------------------------------
// Pass 1: inclusive prefix sum along W. Block = one h row, thread = one d.
// Streaming, perfectly coalesced (1KB per step per block). Unroll x8 for
// outstanding-load ILP; software prefetch one phase ahead (global_prefetch_b8).
// ---------------------------------------------------------------------------
__global__ __launch_bounds__(256) void prefix_w_kernel(const float* __restrict__ feat,
                                                       float* __restrict__ S) {
  const int d = threadIdx.x;
  const int h = blockIdx.x;
  const float* __restrict__ src = feat + (size_t)h * WD + d;
  float* __restrict__ dst = S + (size_t)h * WD + d;

  float acc = 0.0f;
  for (int w = 0; w < WW; w += 8) {
    if (w + 8 < WW) __builtin_prefetch(src + (size_t)(w + 8) * DD, 0, 3);
    float v[8];
#pragma unroll
    for (int i = 0; i < 8; ++i) v[i] = src[(size_t)(w + i) * DD];
#pragma unroll
    for (int i = 0; i < 8; ++i) {
      acc += v[i];
      dst[(size_t)(w + i) * DD] = acc;
    }
  }
}

// ---------------------------------------------------------------------------
// Pass 2: inclusive prefix sum along H, in place on S. Block = one w column,
// thread = one d. Global pattern is 1KB rows at 256KB stride -> ideal for the
// Tensor Data Mover: DMA 32-row x 1KB tiles into LDS, double-buffered, wave 0
// drives TENSORcnt; scan consumes LDS (consecutive d -> conflict-free banks).
// ---------------------------------------------------------------------------
__global__ __launch_bounds__(256) void prefix_h_kernel(float* __restrict__ S) {
  __shared__ float lds[2][HC * DD];   // 2 x 32KB
  const int d = threadIdx.x;
  const int w = blockIdx.x;
  const int NCHUNK = HH / HC;         // 8

  const float* colbase = S + (size_t)w * DD;   // S[0, w, 0]

  if (threadIdx.x < 32) {  // wave 0 only: issue chunk 0
    tdm_load_tile_2d((uint32_t)(uintptr_t)&lds[0][0], colbase,
                     /*tensor_d0=*/DD, /*tensor_d1=*/HC,
                     /*tile_d0=*/DD, /*tile_d1=*/HC,
                     /*stride0=*/(uint64_t)WD);
  }

  float acc = 0.0f;
  for (int c = 0; c < NCHUNK; ++c) {
    if (threadIdx.x < 32) {
      if (c + 1 < NCHUNK) {
        tdm_load_tile_2d((uint32_t)(uintptr_t)&lds[(c + 1) & 1][0],
                         colbase + (size_t)(c + 1) * HC * WD,
                         DD, HC, DD, HC, (uint64_t)WD);
        __builtin_amdgcn_s_wait_tensorcnt(1);   // chunk c landed
      } else {
        __builtin_amdgcn_s_wait_tensorcnt(0);
      }
    }
    __syncthreads();

    const float* __restrict__ buf = &lds[c & 1][0];
    float* __restrict__ dst = S + (size_t)(c * HC) * WD + (size_t)w * DD + d;
#pragma unroll
    for (int hh = 0; hh < HC; ++hh) {
      acc += buf[hh * DD + d];
      dst[(size_t)hh * WD] = acc;
    }
    __syncthreads();   // all reads of buf done before it is re-DMA'd
  }
}

// ---------------------------------------------------------------------------
// Pass 3: per-box lookup. Block = box, thread = channel. Replicates the
// reference bounds math: lo = max(0, floor(f*size)); hi = rint(f*size + 0.5)
// (RNE, matching jnp.round); hi = min(size, max(lo+1, hi)).
// Region sum from 4 SAT corners, divide by pixel count.
// ---------------------------------------------------------------------------
__global__ __launch_bounds__(256) void roi_lookup_kernel(const float* __restrict__ S,
                                                         const float* __restrict__ boxes,
                                                         float* __restrict__ out) {
  const int n = blockIdx.x;
  const int d = threadIdx.x;

  const float x1 = boxes[n * 4 + 0];
  const float y1 = boxes[n * 4 + 1];
  const float x2 = boxes[n * 4 + 2];
  const float y2 = boxes[n * 4 + 3];

  int clo = max(0, (int)floorf(x1 * (float)WW));
  int chi = (int)rintf(x2 * (float)WW + 0.5f);
  chi = min(WW, max(clo + 1, chi));

  int rlo = max(0, (int)floorf(y1 * (float)HH));
  int rhi = (int)rintf(y2 * (float)HH + 0.5f);
  rhi = min(HH, max(rlo + 1, rhi));

  const float* Sd = S + d;
  #define SAT(r, c) Sd[(size_t)(r) * WD + (size_t)(c) * DD]

  float s = SAT(rhi - 1, chi - 1);
  if (rlo > 0)            s -= SAT(rlo - 1, chi - 1);
  if (clo > 0)            s -= SAT(rhi - 1, clo - 1);
  if (rlo > 0 && clo > 0) s += SAT(rlo - 1, clo - 1);
  #undef SAT

  const float cnt = (float)((rhi - rlo) * (chi - clo));
  out[(size_t)n * DD + d] = s / cnt;
}

// ---------------------------------------------------------------------------
extern "C" void kernel_launch(void* const* d_in, const int* in_sizes, int n_in,
                              void* d_out, int out_size, void* d_ws, size_t ws_size,
                              hipStream_t stream) {
  const float* feat  = (const float*)d_in[0];   // (256,256,256) f32
  const float* boxes = (const float*)d_in[1];   // (2048,4) f32
  float* out = (float*)d_out;                   // (2048,256) f32
  float* S   = (float*)d_ws;                    // needs 64MB SAT workspace

  prefix_w_kernel<<<HH, DD, 0, stream>>>(feat, S);
  prefix_h_kernel<<<WW, DD, 0, stream>>>(S);
  roi_lookup_kernel<<<N_BOXES, DD, 0, stream>>>(S, boxes, out);
}